// GAT_gate_31447750542054
// MI455X (gfx1250) — compile-verified
//
#include <hip/hip_runtime.h>
#include <hip/hip_bf16.h>

// Problem constants (match reference)
#define Bn 16
#define Nn 1024
#define Dn 128
#define PCH 8     // softmax chunks per column (128 rows each)

typedef float v2f __attribute__((ext_vector_type(2)));
typedef float v8f __attribute__((ext_vector_type(8)));

// D = A(16x4 f32) * B(4x16 f32) + C(16x16 f32), wave32 WMMA.
// A frag: lane L holds A[M=L%16][K = 2*(L/16) + v], v in {0,1}  -> float2
// B frag: lane L holds B[K = 2*(L/16) + v][N = L%16]
// C/D:    VGPR r, lane L -> C[M = r + 8*(L/16)][N = L%16]
__device__ __forceinline__ v8f wmma4(v2f a, v2f b, v8f c) {
    return __builtin_amdgcn_wmma_f32_16x16x4_f32(
        false, a, false, b, (short)0, c, false, false);
}

// ---------------------------------------------------------------- As = A + A^T
__global__ void k_sym(const float* __restrict__ A, float* __restrict__ As) {
    int t = blockIdx.x * blockDim.x + threadIdx.x;
    if (t < Dn * Dn) {
        int i = t / Dn, j = t % Dn;
        As[t] = A[i * Dn + j] + A[j * Dn + i];
    }
}

// ------------------------------------------------- h = x * Ww^T + Wb  [B*N, D]
// 2x2 tiles: wave computes 32x32 output. B[k][n] = Ww[n][k] -> contiguous f2.
__global__ void k_h(const float* __restrict__ x, const float* __restrict__ Ww,
                    const float* __restrict__ Wb, float* __restrict__ h) {
    const int wave = (blockIdx.x * blockDim.x + threadIdx.x) >> 5; // 2048 waves
    const int lane = threadIdx.x & 31;
    const int l16 = lane & 15, hi = lane >> 4, koff = hi * 2;
    const int mt = wave >> 2;   // 32-row tile over B*N (0..511)
    const int nt = wave & 3;    // 32-col tile over D (0..3)
    const int col0 = nt * 32 + l16, col1 = col0 + 16;
    const float* ap0 = x  + (mt * 32 + l16) * Dn + koff;
    const float* ap1 = ap0 + 16 * Dn;
    const float* bp0 = Ww + col0 * Dn + koff;
    const float* bp1 = Ww + col1 * Dn + koff;
    v8f c00 = {0.f}, c01 = {0.f}, c10 = {0.f}, c11 = {0.f};
#pragma unroll 8
    for (int k = 0; k < Dn; k += 4) {
        v2f a0 = *(const v2f*)(ap0 + k), a1 = *(const v2f*)(ap1 + k);
        v2f b0 = *(const v2f*)(bp0 + k), b1 = *(const v2f*)(bp1 + k);
        c00 = wmma4(a0, b0, c00); c01 = wmma4(a0, b1, c01);
        c10 = wmma4(a1, b0, c10); c11 = wmma4(a1, b1, c11);
    }
    const float bias0 = Wb[col0], bias1 = Wb[col1];
    const int r0 = mt * 32 + 8 * hi, r1 = r0 + 16;
#pragma unroll
    for (int r = 0; r < 8; ++r) {
        h[(r0 + r) * Dn + col0] = c00[r] + bias0;
        h[(r0 + r) * Dn + col1] = c01[r] + bias1;
        h[(r1 + r) * Dn + col0] = c10[r] + bias0;
        h[(r1 + r) * Dn + col1] = c11[r] + bias1;
    }
}

// ------------------------------------------------------- t = h * As  [B*N, D]
// As symmetric -> B[k][n] = As[n][k] (contiguous f2). 2x2 tiles.
__global__ void k_t(const float* __restrict__ h, const float* __restrict__ As,
                    float* __restrict__ t) {
    const int wave = (blockIdx.x * blockDim.x + threadIdx.x) >> 5; // 2048 waves
    const int lane = threadIdx.x & 31;
    const int l16 = lane & 15, hi = lane >> 4, koff = hi * 2;
    const int mt = wave >> 2;
    const int nt = wave & 3;
    const int col0 = nt * 32 + l16, col1 = col0 + 16;
    const float* ap0 = h  + (mt * 32 + l16) * Dn + koff;
    const float* ap1 = ap0 + 16 * Dn;
    const float* bp0 = As + col0 * Dn + koff;
    const float* bp1 = As + col1 * Dn + koff;
    v8f c00 = {0.f}, c01 = {0.f}, c10 = {0.f}, c11 = {0.f};
#pragma unroll 8
    for (int k = 0; k < Dn; k += 4) {
        v2f a0 = *(const v2f*)(ap0 + k), a1 = *(const v2f*)(ap1 + k);
        v2f b0 = *(const v2f*)(bp0 + k), b1 = *(const v2f*)(bp1 + k);
        c00 = wmma4(a0, b0, c00); c01 = wmma4(a0, b1, c01);
        c10 = wmma4(a1, b0, c10); c11 = wmma4(a1, b1, c11);
    }
    const int r0 = mt * 32 + 8 * hi, r1 = r0 + 16;
#pragma unroll
    for (int r = 0; r < 8; ++r) {
        t[(r0 + r) * Dn + col0] = c00[r];
        t[(r0 + r) * Dn + col1] = c01[r];
        t[(r1 + r) * Dn + col0] = c10[r];
        t[(r1 + r) * Dn + col1] = c11[r];
    }
}

// ------------------------------------- e[b,j,k] = sum_l t[b,j,l] * h[b,k,l]
// symmetric scores since t = h*(A+A^T). 2x2 tiles, B frag = h rows (contig f2).
__global__ void k_e(const float* __restrict__ t, const float* __restrict__ h,
                    float* __restrict__ e) {
    const int wave = (blockIdx.x * blockDim.x + threadIdx.x) >> 5; // 16384 waves
    const int lane = threadIdx.x & 31;
    const int l16 = lane & 15, hi = lane >> 4, koff = hi * 2;
    const int b  = wave >> 10;        // 1024 (32x32)-tiles per batch
    const int tb = wave & 1023;
    const int jt = tb >> 5;           // 32-row tile 0..31
    const int kt = tb & 31;           // 32-col tile 0..31
    const float* ap0 = t + (b * Nn + jt * 32 + l16) * Dn + koff;
    const float* ap1 = ap0 + 16 * Dn;
    const float* bp0 = h + (b * Nn + kt * 32 + l16) * Dn + koff;
    const float* bp1 = bp0 + 16 * Dn;
    v8f c00 = {0.f}, c01 = {0.f}, c10 = {0.f}, c11 = {0.f};
#pragma unroll 8
    for (int k = 0; k < Dn; k += 4) {
        v2f a0 = *(const v2f*)(ap0 + k), a1 = *(const v2f*)(ap1 + k);
        v2f b0 = *(const v2f*)(bp0 + k), b1 = *(const v2f*)(bp1 + k);
        c00 = wmma4(a0, b0, c00); c01 = wmma4(a0, b1, c01);
        c10 = wmma4(a1, b0, c10); c11 = wmma4(a1, b1, c11);
    }
    float* ep = e + b * Nn * Nn;
    const int r0 = jt * 32 + 8 * hi, r1 = r0 + 16;
    const int c0 = kt * 32 + l16, c1 = c0 + 16;
#pragma unroll
    for (int r = 0; r < 8; ++r) {
        ep[(r0 + r) * Nn + c0] = c00[r];
        ep[(r0 + r) * Nn + c1] = c01[r];
        ep[(r1 + r) * Nn + c0] = c10[r];
        ep[(r1 + r) * Nn + c1] = c11[r];
    }
}

// -------------------------------- hT[b,d,n] = h[b,n,d] (LDS-tiled transpose)
__global__ void k_tr(const float* __restrict__ h, float* __restrict__ hT) {
    __shared__ float tile[32][33];
    const int bid = blockIdx.x;            // 2048 blocks
    const int b = bid >> 7;
    const int ntile = (bid & 127) >> 2;    // 0..31 over N
    const int dtile = bid & 3;             // 0..3  over D
    const int tx = threadIdx.x, ty = threadIdx.y;  // (32, 8)
#pragma unroll
    for (int r = 0; r < 4; ++r) {
        int n = ntile * 32 + ty + r * 8;
        tile[ty + r * 8][tx] = h[(b * Nn + n) * Dn + dtile * 32 + tx];
    }
    __syncthreads();
#pragma unroll
    for (int r = 0; r < 4; ++r) {
        int dl = ty + r * 8;
        hT[(b * Dn + dtile * 32 + dl) * Nn + ntile * 32 + tx] = tile[tx][dl];
    }
}

// ---- masked softmax over axis j (per column), phase 1: per-chunk online stats
// scores = adj>0 ? e : 0 (masked slots DO feed the denominator with exp(0))
__global__ void k_pmax(const float* __restrict__ adj, const float* __restrict__ e,
                       float* __restrict__ pm, float* __restrict__ ps) {
    const int tid = blockIdx.x * blockDim.x + threadIdx.x; // 0..131071
    const int b  = tid >> 13;
    const int p  = (tid & 8191) >> 10;
    const int kc = tid & 1023;
    const float* ap = adj + (b * Nn + p * 128) * Nn + kc;
    const float* ep = e   + (b * Nn + p * 128) * Nn + kc;
    float m = -3.4e38f, den = 0.f;
    for (int j = 0; j < 128; ++j) {
        const float a = ap[j * Nn];
        const float s = (a > 0.5f) ? ep[j * Nn] : 0.f;
        if (s > m) { den = den * __expf(m - s) + 1.f; m = s; }
        else       { den += __expf(s - m); }
    }
    pm[p * (Bn * Nn) + b * Nn + kc] = m;
    ps[p * (Bn * Nn) + b * Nn + kc] = den;
}

// phase 2: combine chunk stats -> per-column (max, 1/den)
__global__ void k_comb(const float* __restrict__ pm, const float* __restrict__ ps,
                       float* __restrict__ colm, float* __restrict__ colinv) {
    const int tid = blockIdx.x * blockDim.x + threadIdx.x; // 0..16383
    float m = -3.4e38f;
#pragma unroll
    for (int p = 0; p < PCH; ++p) m = fmaxf(m, pm[p * (Bn * Nn) + tid]);
    float den = 0.f;
#pragma unroll
    for (int p = 0; p < PCH; ++p)
        den += ps[p * (Bn * Nn) + tid] * __expf(pm[p * (Bn * Nn) + tid] - m);
    colm[tid] = m;
    colinv[tid] = 1.f / den;
}

// phase 3: attn = adj>0 ? exp(e - m[col]) * inv[col] : 0   (in-place on e)
__global__ void k_attn(const float* __restrict__ adj, float* __restrict__ e,
                       const float* __restrict__ colm, const float* __restrict__ colinv) {
    const int i = (blockIdx.x * blockDim.x + threadIdx.x) * 4; // 16M elems / 4
    const int b  = i >> 20;
    const int kc = i & 1023;
    const float4 ev = *(const float4*)(e + i);
    const float4 av = *(const float4*)(adj + i);
    const float4 mv = *(const float4*)(colm + b * Nn + kc);
    const float4 iv = *(const float4*)(colinv + b * Nn + kc);
    float4 o;
    o.x = (av.x > 0.5f) ? __expf(ev.x - mv.x) * iv.x : 0.f;
    o.y = (av.y > 0.5f) ? __expf(ev.y - mv.y) * iv.y : 0.f;
    o.z = (av.z > 0.5f) ? __expf(ev.z - mv.z) * iv.z : 0.f;
    o.w = (av.w > 0.5f) ? __expf(ev.w - mv.w) * iv.w : 0.f;
    *(float4*)(e + i) = o;
}

// ----------------- h' = relu(attn * h): 2x2 tiles, B frags from hT (contig f2)
__global__ void k_hp(const float* __restrict__ attn, const float* __restrict__ hT,
                     float* __restrict__ hp) {
    const int wave = (blockIdx.x * blockDim.x + threadIdx.x) >> 5; // 2048 waves
    const int lane = threadIdx.x & 31;
    const int l16 = lane & 15, hi = lane >> 4, koff = hi * 2;
    const int b  = wave >> 7;         // 128 tiles per batch
    const int tb = wave & 127;
    const int it = tb >> 2;           // 32-row tile 0..31
    const int nt = tb & 3;            // 32-col tile over D (0..3)
    const float* ap0 = attn + (b * Nn + it * 32 + l16) * Nn + koff;
    const float* ap1 = ap0 + 16 * Nn;
    const float* bp0 = hT + (b * Dn + nt * 32 + l16) * Nn + koff;
    const float* bp1 = bp0 + 16 * Nn;
    v8f c00 = {0.f}, c01 = {0.f}, c10 = {0.f}, c11 = {0.f};
    for (int kk = 0; kk < Nn; kk += 256) {
        // pull the next 1KB chunk of each stream toward the WGP (global_prefetch_b8)
        __builtin_prefetch(ap0 + kk + 256, 0, 1);
        __builtin_prefetch(ap1 + kk + 256, 0, 1);
        __builtin_prefetch(bp0 + kk + 256, 0, 1);
        __builtin_prefetch(bp1 + kk + 256, 0, 1);
#pragma unroll 8
        for (int k = kk; k < kk + 256; k += 4) {
            v2f a0 = *(const v2f*)(ap0 + k), a1 = *(const v2f*)(ap1 + k);
            v2f b0 = *(const v2f*)(bp0 + k), b1 = *(const v2f*)(bp1 + k);
            c00 = wmma4(a0, b0, c00); c01 = wmma4(a0, b1, c01);
            c10 = wmma4(a1, b0, c10); c11 = wmma4(a1, b1, c11);
        }
    }
    const int r0 = b * Nn + it * 32 + 8 * hi, r1 = r0 + 16;
    const int c0 = nt * 32 + l16, c1 = c0 + 16;
#pragma unroll
    for (int r = 0; r < 8; ++r) {
        hp[(r0 + r) * Dn + c0] = fmaxf(c00[r], 0.f);
        hp[(r0 + r) * Dn + c1] = fmaxf(c01[r], 0.f);
        hp[(r1 + r) * Dn + c0] = fmaxf(c10[r], 0.f);
        hp[(r1 + r) * Dn + c1] = fmaxf(c11[r], 0.f);
    }
}

// ----------------- per-node gates + output. One wave32 per node, 4 feats/lane.
__global__ void k_gates(const float* __restrict__ x, const float* __restrict__ hp,
                        const float* __restrict__ wi_u, const float* __restrict__ wi_x,
                        const float* __restrict__ wf_u, const float* __restrict__ wf_x,
                        const float* __restrict__ wo_u, const float* __restrict__ wo_x,
                        float* __restrict__ out) {
    const int node = (blockIdx.x * blockDim.x + threadIdx.x) >> 5; // 0..16383
    const int lane = threadIdx.x & 31;
    const int f = lane * 4;
    const float4 h4 = *(const float4*)(hp + node * Dn + f);
    const float4 x4 = *(const float4*)(x  + node * Dn + f);
    const float4 iu = *(const float4*)(wi_u + f);
    const float4 ix = *(const float4*)(wi_x + f);
    const float4 fu = *(const float4*)(wf_u + f);
    const float4 fx = *(const float4*)(wf_x + f);
    const float4 ou = *(const float4*)(wo_u + f);
    const float4 ox = *(const float4*)(wo_x + f);
    float pi = h4.x*iu.x + h4.y*iu.y + h4.z*iu.z + h4.w*iu.w
             + x4.x*ix.x + x4.y*ix.y + x4.z*ix.z + x4.w*ix.w;
    float pf = h4.x*fu.x + h4.y*fu.y + h4.z*fu.z + h4.w*fu.w
             + x4.x*fx.x + x4.y*fx.y + x4.z*fx.z + x4.w*fx.w;
    float po = h4.x*ou.x + h4.y*ou.y + h4.z*ou.z + h4.w*ou.w
             + x4.x*ox.x + x4.y*ox.y + x4.z*ox.z + x4.w*ox.w;
#pragma unroll
    for (int m = 16; m >= 1; m >>= 1) {
        pi += __shfl_xor(pi, m, 32);
        pf += __shfl_xor(pf, m, 32);
        po += __shfl_xor(po, m, 32);
    }
    const float ic = 1.f / (1.f + __expf(-pi));
    const float fc = 1.f / (1.f + __expf(-pf));
    const float oc = 1.f / (1.f + __expf(-po));
    float4 o;
    o.x = oc * tanhf(ic * h4.x + fc * x4.x);
    o.y = oc * tanhf(ic * h4.y + fc * x4.y);
    o.z = oc * tanhf(ic * h4.z + fc * x4.z);
    o.w = oc * tanhf(ic * h4.w + fc * x4.w);
    *(float4*)(out + node * Dn + f) = o;
}

extern "C" void kernel_launch(void* const* d_in, const int* in_sizes, int n_in,
                              void* d_out, int out_size, void* d_ws, size_t ws_size,
                              hipStream_t stream) {
    (void)in_sizes; (void)n_in; (void)out_size; (void)ws_size;
    const float* x    = (const float*)d_in[0];   // [B,N,D]
    const float* adj  = (const float*)d_in[1];   // [B,N,N]
    const float* Ww   = (const float*)d_in[2];   // [D,D]
    const float* Wb   = (const float*)d_in[3];   // [D]
    const float* A    = (const float*)d_in[4];   // [D,D]
    const float* wi_u = (const float*)d_in[5];
    const float* wi_x = (const float*)d_in[6];
    const float* wf_u = (const float*)d_in[7];
    const float* wf_x = (const float*)d_in[8];
    const float* wo_u = (const float*)d_in[9];
    const float* wo_x = (const float*)d_in[10];
    float* out = (float*)d_out;

    float* ws  = (float*)d_ws;
    float* h   = ws;                   // B*N*D = 2,097,152 f
    float* t   = ws + 2097152;         // B*N*D   (recycled as hT after k_e)
    float* e   = ws + 4194304;         // B*N*N = 16,777,216 f (scores -> attn)
    float* As  = ws + 20971520;        // D*D = 16,384 f
    float* hp  = ws + 20987904;        // B*N*D
    float* pm  = ws + 23085056;        // 16384*PCH = 131,072 f
    float* psm = ws + 23216128;        // 131,072 f
    float* cm  = ws + 23347200;        // 16,384 f
    float* ci  = ws + 23363584;        // 16,384 f  (end: 23,379,968 f ~ 93.5 MB)

    k_sym  <<<   64, 256, 0, stream>>>(A, As);
    k_h    <<<  256, 256, 0, stream>>>(x, Ww, Wb, h);      // 2048 waves, 2x2 tiles
    k_t    <<<  256, 256, 0, stream>>>(h, As, t);          // 2048 waves
    k_e    <<< 2048, 256, 0, stream>>>(t, h, e);           // 16384 waves
    k_tr   <<< 2048, dim3(32, 8), 0, stream>>>(h, t);      // t := hT
    k_pmax <<<  512, 256, 0, stream>>>(adj, e, pm, psm);   // 131072 threads
    k_comb <<<   64, 256, 0, stream>>>(pm, psm, cm, ci);   // 16384 threads
    k_attn <<<16384, 256, 0, stream>>>(adj, e, cm, ci);    // 4M threads, float4
    k_hp   <<<  256, 256, 0, stream>>>(e, t, hp);          // 2048 waves, 2x2 tiles
    k_gates<<< 2048, 256, 0, stream>>>(x, hp, wi_u, wi_x, wf_u, wf_x,
                                       wo_u, wo_x, out);   // 16384 waves
}